// ExpertChoiceMoEMatcher_74105365725282
// MI455X (gfx1250) — compile-verified
//
#include <hip/hip_runtime.h>
#include <hip/hip_bf16.h>
#include <math.h>

// ---------------------------------------------------------------------------
// ExpertChoice MoE with complex experts, CDNA5 (gfx1250, wave32).
// Dominant work: 16 experts x complex GEMM [2048x512]x[512x512] (~69 GFLOP
// fp32) -> V_WMMA_F32_16X16X4_F32, with double-buffered LDS staging via
// GLOBAL_LOAD_ASYNC_TO_LDS_B128 (ASYNCcnt) when available.
// ---------------------------------------------------------------------------

#define B_T   32768
#define DD    512
#define EE    16
#define KNODE 2048   // B_T / EE

typedef __attribute__((ext_vector_type(2))) float v2f;
typedef __attribute__((ext_vector_type(8))) float v8f;
typedef __attribute__((ext_vector_type(4))) int   v4i;

#if defined(__gfx1250__) && __has_builtin(__builtin_amdgcn_global_load_async_to_lds_b128)
#define HAVE_ASYNC_LDS 1
#else
#define HAVE_ASYNC_LDS 0
#endif

// global(16B) -> LDS(16B) copy; async (no VGPR round-trip) when available.
__device__ __forceinline__ void cp16(const float* __restrict__ g, float* l) {
#if HAVE_ASYNC_LDS
  __builtin_amdgcn_global_load_async_to_lds_b128(
      (__attribute__((address_space(1))) v4i*)g,
      (__attribute__((address_space(3))) v4i*)l, 0, 0);
#else
  *(float4*)l = *(const float4*)g;
#endif
}

template <int N>
__device__ __forceinline__ void wait_async() {
#if HAVE_ASYNC_LDS
#if __has_builtin(__builtin_amdgcn_s_wait_asynccnt)
  __builtin_amdgcn_s_wait_asynccnt(N);
#else
  asm volatile("s_wait_asynccnt %0" ::"n"(N));
#endif
#endif
}

__device__ __forceinline__ unsigned fkey(float f) {
  unsigned u = __float_as_uint(f);
  return (u & 0x80000000u) ? ~u : (u | 0x80000000u);
}

// -------------------------------------------------------------------- gating
__global__ __launch_bounds__(256) void gate_kernel(
    const float* __restrict__ x, const float* __restrict__ gw,
    float* __restrict__ scores) {
  const int wave = blockIdx.x * 8 + (threadIdx.x >> 5);
  const int lane = threadIdx.x & 31;
  const float* xr = x + (size_t)wave * 1024;
  float acc[EE];
#pragma unroll
  for (int e = 0; e < EE; ++e) acc[e] = 0.f;
  for (int d = lane; d < 1024; d += 32) {
    float xv = xr[d];
    const float* g = gw + d * EE;
#pragma unroll
    for (int e = 0; e < EE; ++e) acc[e] += xv * g[e];
  }
#pragma unroll
  for (int e = 0; e < EE; ++e) {
    float v = acc[e];
#pragma unroll
    for (int off = 16; off > 0; off >>= 1) v += __shfl_xor(v, off, 32);
    if (lane == 0) scores[(size_t)wave * EE + e] = v;
  }
}

// ---------------------------------------------------------------------- zero
__global__ void zero_kernel(float4* __restrict__ res4, float* __restrict__ counts) {
  size_t i = (size_t)blockIdx.x * blockDim.x + threadIdx.x;
  size_t stride = (size_t)gridDim.x * blockDim.x;
  for (size_t j = i; j < (size_t)B_T * DD * 2 / 4; j += stride)
    res4[j] = make_float4(0.f, 0.f, 0.f, 0.f);
  for (size_t j = i; j < B_T; j += stride) counts[j] = 0.f;
}

// ------------------------------------------------------- top-k (radix select)
__global__ __launch_bounds__(1024) void topk_kernel(
    const float* __restrict__ scores, int* __restrict__ sel_idx,
    float* __restrict__ sel_score, float* __restrict__ counts) {
  const int e = blockIdx.x;
  const int tid = threadIdx.x;
  __shared__ unsigned hist[256];
  __shared__ unsigned s_prefix, s_want, s_gt, s_eq;
  if (tid == 0) { s_prefix = 0u; s_want = KNODE; s_gt = 0u; s_eq = 0u; }
  __syncthreads();

  for (int pass = 0; pass < 4; ++pass) {
    const int shift = 24 - 8 * pass;
    if (tid < 256) hist[tid] = 0u;
    __syncthreads();
    const unsigned maskhi = pass ? (0xFFFFFFFFu << (shift + 8)) : 0u;
    const unsigned pref = s_prefix;
    for (int t = tid; t < B_T; t += 1024) {
      unsigned key = fkey(scores[(size_t)t * EE + e]);
      if ((key & maskhi) == pref) atomicAdd(&hist[(key >> shift) & 255u], 1u);
    }
    __syncthreads();
    if (tid == 0) {
      unsigned want = s_want;
      for (int b = 255; b >= 0; --b) {
        unsigned c = hist[b];
        if (c >= want) { s_prefix = pref | ((unsigned)b << shift); break; }
        want -= c;
      }
      s_want = want;
    }
    __syncthreads();
  }

  const unsigned T = s_prefix;
  const unsigned wf = s_want;
  const unsigned gt_total = KNODE - wf;
  for (int t = tid; t < B_T; t += 1024) {
    float sc = scores[(size_t)t * EE + e];
    unsigned key = fkey(sc);
    int slot = -1;
    if (key > T) {
      slot = (int)atomicAdd(&s_gt, 1u);
    } else if (key == T) {
      unsigned ep = atomicAdd(&s_eq, 1u);
      if (ep < wf) slot = (int)(gt_total + ep);
    }
    if (slot >= 0) {
      sel_idx[e * KNODE + slot] = t;
      sel_score[e * KNODE + slot] = sc;
      atomicAdd(&counts[t], 1.0f);
    }
  }
}

// --------------------------------------------------- sort top-k (bitonic, LDS)
__global__ __launch_bounds__(1024) void sort_kernel(
    const float* __restrict__ scores, const int* __restrict__ sel_idx,
    float* __restrict__ out_idx, float* __restrict__ out_sc) {
  const int e = blockIdx.x;
  const int tid = threadIdx.x;
  __shared__ unsigned long long keys[KNODE];
  for (int i = tid; i < KNODE; i += 1024) {
    int t = sel_idx[e * KNODE + i];
    unsigned k32 = fkey(scores[(size_t)t * EE + e]);
    keys[i] = ((unsigned long long)k32 << 32) | (unsigned long long)(0xFFFFFFFFu - (unsigned)t);
  }
  __syncthreads();
  for (int k2 = 2; k2 <= KNODE; k2 <<= 1) {
    for (int j = k2 >> 1; j > 0; j >>= 1) {
      for (int i = tid; i < KNODE; i += 1024) {
        int ixj = i ^ j;
        if (ixj > i) {
          unsigned long long a = keys[i], b = keys[ixj];
          bool desc = (i & k2) == 0;
          if (desc ? (a < b) : (a > b)) { keys[i] = b; keys[ixj] = a; }
        }
      }
      __syncthreads();
    }
  }
  for (int i = tid; i < KNODE; i += 1024) {
    unsigned t = 0xFFFFFFFFu - (unsigned)(keys[i] & 0xFFFFFFFFu);
    out_idx[(size_t)i * EE + e] = (float)t;
    out_sc[(size_t)i * EE + e] = scores[(size_t)t * EE + e];
  }
}

// --------------------------------------------------------- expert GEMM (WMMA)
// Block: 256 threads (8 waves). Output tile: M=32 tokens x N=128. K staged in
// chunks of 16 complex through double-buffered LDS (async copies). Each wave
// (mw,nw) owns a 16x32 slab = two 16x16 subtiles with yr/yi accumulators.
#define KC 16
#define ABUF (32 * KC * 2)   // 1024 floats / buffer
#define BBUF (KC * 128 * 2)  // 4096 floats / buffer

__global__ __launch_bounds__(256) void moe_gemm_kernel(
    const float* __restrict__ x, const float* __restrict__ w,
    const int* __restrict__ sel_idx, const float* __restrict__ sel_score,
    float* __restrict__ out) {
  __shared__ __align__(16) float ldsA[2 * ABUF];  //  8 KB
  __shared__ __align__(16) float ldsB[2 * BBUF];  // 32 KB
  __shared__ int   s_tok[32];
  __shared__ float s_sc[32];

  const int mtile = blockIdx.x;   // 0..63
  const int ntile = blockIdx.y;   // 0..3
  const int e     = blockIdx.z;   // 0..15
  const int tid   = threadIdx.x;
  const int lane  = tid & 31;
  const int wave  = tid >> 5;
  const int mw = wave & 1, nw = wave >> 1;
  const int lo = lane & 15, hi = lane >> 4;

  if (tid < 32) {
    int slot = e * KNODE + mtile * 32 + tid;
    s_tok[tid] = sel_idx[slot];
    s_sc[tid]  = sel_score[slot];
  }
  __syncthreads();

  const int fbase = ntile * 128;
  const size_t wexp = (size_t)e * 524288;  // e*512*512*2

  // staging addresses (per thread, per chunk):
  //  A: 32 rows x 8 float4  -> 256 float4, 1/thread
  const int a_row = tid >> 3, a_c4 = tid & 7;
  const float* a_src0 = x + (size_t)s_tok[a_row] * 1024 + a_c4 * 4;
  //  B: 16 rows x 64 float4 -> 1024 float4, 4/thread
  const int b_row = tid >> 6, b_c4 = tid & 63;
  const float* b_src0 = w + wexp + (size_t)b_row * 1024 + fbase * 2 + b_c4 * 4;

  auto stage = [&](int chunk, int buf) {
    const int kb = chunk * KC;
    cp16(a_src0 + kb * 2, &ldsA[buf * ABUF + a_row * (KC * 2) + a_c4 * 4]);
#pragma unroll
    for (int it = 0; it < 4; ++it) {
      int row = b_row + it * 4;
      cp16(b_src0 + (size_t)(kb + it * 4) * 1024,
           &ldsB[buf * BBUF + row * 256 + b_c4 * 4]);
    }
  };

  v8f yr0 = {}, yi0 = {}, yr1 = {}, yi1 = {};
  stage(0, 0);

  const int nchunks = DD / KC;  // 32
  for (int c = 0; c < nchunks; ++c) {
    const int cur = c & 1;
    if (c + 1 < nchunks) {
      stage(c + 1, cur ^ 1);
      wait_async<5>();   // 5 in flight for next chunk; current chunk landed
    } else {
      wait_async<0>();
    }
    __syncthreads();     // all waves' stages for chunk c visible

    const float* A = &ldsA[cur * ABUF];
    const float* B = &ldsB[cur * BBUF];
    const int arow = (16 * mw + lo) * (KC * 2);
#pragma unroll
    for (int kl = 0; kl < KC; kl += 4) {
      float4 fa = *(const float4*)&A[arow + (kl + 2 * hi) * 2];
      v2f ar = {fa.x, fa.z};
      v2f ai = {fa.y, fa.w};
      v2f an = {-fa.y, -fa.w};
      {
        const int fb = nw * 32;
        float2 b0 = *(const float2*)&B[(kl + 2 * hi) * 256 + (fb + lo) * 2];
        float2 b1 = *(const float2*)&B[(kl + 2 * hi + 1) * 256 + (fb + lo) * 2];
        v2f br = {b0.x, b1.x}, bi = {b0.y, b1.y};
        yr0 = __builtin_amdgcn_wmma_f32_16x16x4_f32(false, ar, false, br, (short)0, yr0, false, false);
        yr0 = __builtin_amdgcn_wmma_f32_16x16x4_f32(false, an, false, bi, (short)0, yr0, false, false);
        yi0 = __builtin_amdgcn_wmma_f32_16x16x4_f32(false, ar, false, bi, (short)0, yi0, false, false);
        yi0 = __builtin_amdgcn_wmma_f32_16x16x4_f32(false, ai, false, br, (short)0, yi0, false, false);
      }
      {
        const int fb = nw * 32 + 16;
        float2 b0 = *(const float2*)&B[(kl + 2 * hi) * 256 + (fb + lo) * 2];
        float2 b1 = *(const float2*)&B[(kl + 2 * hi + 1) * 256 + (fb + lo) * 2];
        v2f br = {b0.x, b1.x}, bi = {b0.y, b1.y};
        yr1 = __builtin_amdgcn_wmma_f32_16x16x4_f32(false, ar, false, br, (short)0, yr1, false, false);
        yr1 = __builtin_amdgcn_wmma_f32_16x16x4_f32(false, an, false, bi, (short)0, yr1, false, false);
        yi1 = __builtin_amdgcn_wmma_f32_16x16x4_f32(false, ar, false, bi, (short)0, yi1, false, false);
        yi1 = __builtin_amdgcn_wmma_f32_16x16x4_f32(false, ai, false, br, (short)0, yi1, false, false);
      }
    }
    __syncthreads();     // readers done before buffer `cur` is re-staged
  }

  // epilogue: weight by gate score, scatter-add into accumulation buffer
#pragma unroll
  for (int r = 0; r < 8; ++r) {
    int m = r + 8 * hi;
    int row = 16 * mw + m;
    float s = s_sc[row];
    size_t gbase = (size_t)s_tok[row] * 1024;
    int f0 = fbase + nw * 32 + lo;
    atomicAdd(out + gbase + (size_t)f0 * 2,     yr0[r] * s);
    atomicAdd(out + gbase + (size_t)f0 * 2 + 1, yi0[r] * s);
    int f1 = f0 + 16;
    atomicAdd(out + gbase + (size_t)f1 * 2,     yr1[r] * s);
    atomicAdd(out + gbase + (size_t)f1 * 2 + 1, yi1[r] * s);
  }
}

// ------------------------------------------------------------------ finalize
__global__ __launch_bounds__(256) void finalize_kernel(
    float* __restrict__ res, const float* __restrict__ counts,
    const float* __restrict__ bias, float* __restrict__ out_counts) {
  size_t i = (size_t)blockIdx.x * blockDim.x + threadIdx.x;
  int t = (int)(i >> 9);
  int d = (int)(i & 511);
  float c = counts[t];
  float denom = fmaxf(c, 1.0f);
  float re = res[(size_t)t * 1024 + d * 2]     / denom;
  float im = res[(size_t)t * 1024 + d * 2 + 1] / denom;
  float mag = sqrtf(re * re + im * im);
  float safe = fmaxf(mag, 1e-8f);
  float act = fmaxf(mag + bias[d], 0.0f);
  float s = act / safe;
  res[(size_t)t * 1024 + d * 2]     = re * s;
  res[(size_t)t * 1024 + d * 2 + 1] = im * s;
  if (d == 0) out_counts[t] = c;
}

// ---------------------------------------------------------------------------
extern "C" void kernel_launch(void* const* d_in, const int* in_sizes, int n_in,
                              void* d_out, int out_size, void* d_ws, size_t ws_size,
                              hipStream_t stream) {
  const float* x    = (const float*)d_in[0];   // [32768,512,2]
  const float* gw   = (const float*)d_in[1];   // [1024,16]
  const float* wexp = (const float*)d_in[2];   // [16,512,512,2]
  const float* bias = (const float*)d_in[3];   // [512]

  float* out = (float*)d_out;
  const size_t RES_N = (size_t)B_T * DD * 2;           // 33,554,432
  float* out_idx    = out + RES_N;                     // [2048,16]
  float* out_scores = out_idx + (size_t)KNODE * EE;    // [2048,16]
  float* out_counts = out_scores + (size_t)KNODE * EE; // [32768]

  char* ws = (char*)d_ws;
  float* scores    = (float*)ws;                               // 2 MB
  int*   sel_idx   = (int*)(ws + 2097152);                     // 128 KB
  float* sel_score = (float*)(ws + 2097152 + 131072);          // 128 KB
  float* counts    = (float*)(ws + 2097152 + 2 * 131072);      // 128 KB

  gate_kernel<<<B_T / 8, 256, 0, stream>>>(x, gw, scores);
  zero_kernel<<<4096, 256, 0, stream>>>((float4*)out, counts);
  topk_kernel<<<EE, 1024, 0, stream>>>(scores, sel_idx, sel_score, counts);
  sort_kernel<<<EE, 1024, 0, stream>>>(scores, sel_idx, out_idx, out_scores);
  moe_gemm_kernel<<<dim3(KNODE / 32, DD / 128, EE), 256, 0, stream>>>(
      x, wexp, sel_idx, sel_score, out);
  finalize_kernel<<<(B_T * DD) / 256, 256, 0, stream>>>(out, counts, bias, out_counts);
}